// Quantizer_32418413150309
// MI455X (gfx1250) — compile-verified
//
#include <hip/hip_runtime.h>
#include <hip/hip_bf16.h>

typedef __attribute__((ext_vector_type(2))) float v2f;
typedef __attribute__((ext_vector_type(8))) float v8f;

#define K_CODES     512
#define C_CH        64
#define NPOS        131072        // 4*32*32*32
#define CSTRIDE     32768         // elements between channel planes
#define BSTRIDE     2097152       // 64*32768 elements per batch
#define EPAD        68            // padded LDS row stride (floats), keeps b64 reads conflict-free
#define NWG         256
#define TPB         256
#define WAVES_PER_WG 8
#define TOTAL_WAVES (NWG * WAVES_PER_WG)   // 2048
#define PAIRS       (NPOS / 32)            // 4096 pairs of 16-row M-tiles

__launch_bounds__(TPB, 1)
__global__ void vq_main(const float* __restrict__ in,
                        const float* __restrict__ embed,
                        float* __restrict__ qst,
                        float* __restrict__ idxf,
                        float* __restrict__ wspart)
{
    __shared__ float lds_e[K_CODES * EPAD];   // 139264 B codebook, padded
    __shared__ float lds_e2[K_CODES];         // 0.5*||e||^2
    __shared__ int   lds_idx[WAVES_PER_WG][32];
    __shared__ float lds_red[WAVES_PER_WG];

    const int tid  = threadIdx.x;
    const int wave = tid >> 5;
    const int lane = tid & 31;
    const int nl   = lane & 15;   // column / row-in-halfwave
    const int hiK  = lane >> 4;   // which K-subpair this half-wave holds

    // ---- cooperative codebook load into LDS (coalesced) ----
    for (int t = tid; t < K_CODES * C_CH; t += TPB) {
        int n = t >> 6, c = t & 63;
        lds_e[n * EPAD + c] = embed[t];
    }
    __syncthreads();
    for (int n = tid; n < K_CODES; n += TPB) {
        float s = 0.f;
        #pragma unroll 8
        for (int c = 0; c < C_CH; ++c) { float v = lds_e[n * EPAD + c]; s += v * v; }
        lds_e2[n] = 0.5f * s;
    }
    __syncthreads();

    float acc = 0.f;
    const int gwave = blockIdx.x * WAVES_PER_WG + wave;

    for (int p = gwave; p < PAIRS; p += TOTAL_WAVES) {
        const int base = p * 32;

        // ---- A fragments: two 16x64 fp32 tiles held in registers ----
        // A layout (16x4 f32): lanes 0-15 rows, v0 = K0(low)/K2(high), v1 = K1/K3
        v2f A0[16], A1[16];
        {
            int pos0 = base + nl;
            const float* xp0 = in + (pos0 >> 15) * BSTRIDE + (pos0 & 32767);
            int pos1 = base + 16 + nl;
            const float* xp1 = in + (pos1 >> 15) * BSTRIDE + (pos1 & 32767);
            #pragma unroll
            for (int kk = 0; kk < 16; ++kk) {
                int c0 = kk * 4 + hiK * 2;
                v2f a; a.x = xp0[c0 * CSTRIDE]; a.y = xp0[(c0 + 1) * CSTRIDE];
                A0[kk] = a;
                v2f b; b.x = xp1[c0 * CSTRIDE]; b.y = xp1[(c0 + 1) * CSTRIDE];
                A1[kk] = b;
            }
        }

        float bv0[8], bv1[8]; int bi0[8], bi1[8];
        #pragma unroll
        for (int j = 0; j < 8; ++j) { bv0[j] = -1e30f; bv1[j] = -1e30f; bi0[j] = 0; bi1[j] = 0; }

        // ---- sweep 32 code tiles; 16 chained f32 WMMAs per tile per M-tile ----
        for (int nt = 0; nt < 32; ++nt) {
            v8f c0 = {}; v8f c1 = {};
            const int nrow = nt * 16 + nl;
            #pragma unroll
            for (int kk = 0; kk < 16; ++kk) {
                int cch = kk * 4 + hiK * 2;
                v2f bfrag = *(const v2f*)(lds_e + nrow * EPAD + cch);  // ds_load_b64
                c0 = __builtin_amdgcn_wmma_f32_16x16x4_f32(false, A0[kk], false, bfrag,
                                                           (short)0, c0, false, false);
                c1 = __builtin_amdgcn_wmma_f32_16x16x4_f32(false, A1[kk], false, bfrag,
                                                           (short)0, c1, false, false);
            }
            float e2 = lds_e2[nrow];
            #pragma unroll
            for (int j = 0; j < 8; ++j) {
                float s0 = c0[j] - e2;
                if (s0 > bv0[j]) { bv0[j] = s0; bi0[j] = nrow; }
                float s1 = c1[j] - e2;
                if (s1 > bv1[j]) { bv1[j] = s1; bi1[j] = nrow; }
            }
        }

        // ---- argmax across the 16 lanes of each half (wave32 butterfly) ----
        #pragma unroll
        for (int j = 0; j < 8; ++j) {
            float v0 = bv0[j]; int i0 = bi0[j];
            float v1 = bv1[j]; int i1 = bi1[j];
            #pragma unroll
            for (int off = 8; off > 0; off >>= 1) {
                float vo = __shfl_xor(v0, off, 16); int io = __shfl_xor(i0, off, 16);
                if (vo > v0 || (vo == v0 && io < i0)) { v0 = vo; i0 = io; }
                vo = __shfl_xor(v1, off, 16); io = __shfl_xor(i1, off, 16);
                if (vo > v1 || (vo == v1 && io < i1)) { v1 = vo; i1 = io; }
            }
            if (nl == 0) {                     // lane 0 -> rows 0-7, lane 16 -> rows 8-15
                int rowoff = j + hiK * 8;
                lds_idx[wave][rowoff]      = i0;
                lds_idx[wave][16 + rowoff] = i1;
            }
        }
        asm volatile("s_wait_dscnt 0" ::: "memory");

        // ---- gather codebook row, write quantized_st + index, accumulate loss ----
        int myidx = lds_idx[wave][lane];
        int pos = base + lane;
        const float* xp = in  + (pos >> 15) * BSTRIDE + (pos & 32767);
        float*       qp = qst + (pos >> 15) * BSTRIDE + (pos & 32767);
        const float* ep = lds_e + myidx * EPAD;
        #pragma unroll 8
        for (int c = 0; c < C_CH; ++c) {
            float q = ep[c];
            float x = xp[c * CSTRIDE];
            float d = q - x;
            qp[c * CSTRIDE] = x + d;   // straight-through estimator, reference order
            acc += d * d;
        }
        idxf[pos] = (float)myidx;
    }

    // ---- deterministic per-workgroup loss partial ----
    #pragma unroll
    for (int off = 16; off > 0; off >>= 1) acc += __shfl_xor(acc, off, 32);
    if (lane == 0) lds_red[wave] = acc;
    __syncthreads();
    if (tid == 0) {
        float s = 0.f;
        for (int w = 0; w < WAVES_PER_WG; ++w) s += lds_red[w];
        wspart[blockIdx.x] = s;
    }
}

__global__ void vq_finalize(const float* __restrict__ wspart, float* __restrict__ out_loss)
{
    __shared__ float sm[NWG];
    int t = threadIdx.x;
    sm[t] = wspart[t];
    __syncthreads();
    for (int s = NWG / 2; s > 0; s >>= 1) {
        if (t < s) sm[t] += sm[t + s];
        __syncthreads();
    }
    if (t == 0) out_loss[0] = 0.25f * sm[0] / 8388608.0f;
}

extern "C" void kernel_launch(void* const* d_in, const int* in_sizes, int n_in,
                              void* d_out, int out_size, void* d_ws, size_t ws_size,
                              hipStream_t stream)
{
    (void)in_sizes; (void)n_in; (void)out_size; (void)ws_size;
    const float* in    = (const float*)d_in[0];   // [4,64,32,32,32]
    const float* embed = (const float*)d_in[1];   // [512,64]
    float* out   = (float*)d_out;                 // [loss | quantized_st | indices]
    float* qst   = out + 1;
    float* idxf  = out + 1 + 8388608;
    float* wsp   = (float*)d_ws;                  // 256 partials

    vq_main<<<NWG, TPB, 0, stream>>>(in, embed, qst, idxf, wsp);
    vq_finalize<<<1, NWG, 0, stream>>>(wsp, out);
}